// BSplineBasis_26207890440653
// MI455X (gfx1250) — compile-verified
//
#include <hip/hip_runtime.h>
#include <math.h>

#define N_POINTS 500000
#define NB 60            // num_knots - degree - 1 = 64 - 3 - 1
#define NKNOTS 64
#define PAIRS_PER_POINT (NB / 2)                    // 30
#define TOTAL_PAIRS (N_POINTS * PAIRS_PER_POINT)    // 15,000,000 (fits in int)

// ---- monotone encoding of f32 so unsigned min/max == float min/max ----
__device__ __forceinline__ unsigned f32_enc(float f) {
    unsigned u = __float_as_uint(f);
    return (u & 0x80000000u) ? ~u : (u | 0x80000000u);
}
__device__ __forceinline__ float f32_dec(unsigned u) {
    unsigned v = (u & 0x80000000u) ? (u ^ 0x80000000u) : ~u;
    return __uint_as_float(v);
}

__global__ void bspline_init_ws(unsigned* ws) {
    if (threadIdx.x == 0) {
        ws[0] = 0xFFFFFFFFu;  // min identity (encoded)
        ws[1] = 0x00000000u;  // max identity (encoded)
    }
}

// Exact global min/max reduction: wave32 shuffle -> LDS atomics -> global atomics.
__global__ __launch_bounds__(256) void bspline_minmax(const float* __restrict__ x,
                                                      unsigned* __restrict__ ws, int n) {
    __shared__ unsigned smn, smx;
    if (threadIdx.x == 0) { smn = 0xFFFFFFFFu; smx = 0u; }
    __syncthreads();

    unsigned lmn = 0xFFFFFFFFu, lmx = 0u;
    for (int idx = blockIdx.x * blockDim.x + threadIdx.x; idx < n;
         idx += gridDim.x * blockDim.x) {
        unsigned e = f32_enc(x[idx]);
        lmn = min(lmn, e);
        lmx = max(lmx, e);
    }
    // wave32 butterfly reduction
    for (int off = 16; off > 0; off >>= 1) {
        lmn = min(lmn, (unsigned)__shfl_xor((int)lmn, off, 32));
        lmx = max(lmx, (unsigned)__shfl_xor((int)lmx, off, 32));
    }
    if ((threadIdx.x & 31) == 0) {
        atomicMin(&smn, lmn);   // ds_min_u32
        atomicMax(&smx, lmx);   // ds_max_u32
    }
    __syncthreads();
    if (threadIdx.x == 0) {
        atomicMin(&ws[0], smn); // global_atomic_min_u32
        atomicMax(&ws[1], smx);
    }
}

// Evaluate one basis element: seg = {s0..s4}, cnt = #{m : seg[m] <= xn}.
// Faithful to reference: ell = clamp(2+cnt, 3, 6); padded T; de Boor with
// den==0 -> w=0 guard; result = res[6-ell]; nan_to_num.
__device__ __forceinline__ float bspline_eval_elem(float s0, float s1, float s2,
                                                   float s3, float s4,
                                                   float xn, int cnt) {
    int ell = 2 + cnt;
    ell = ell < 3 ? 3 : (ell > 6 ? 6 : ell);

    // T[idx], idx in [1..9]: idx<=2 -> s0-1 ; 3..7 -> seg[idx-3] ; idx>=8 -> s4+1
    auto Tat = [&](int idx) -> float {
        float r = (idx <= 3) ? s0 : (idx == 4) ? s1 : (idx == 5) ? s2
                  : (idx == 6) ? s3 : s4;
        if (idx <= 2) r = s0 - 1.0f;
        if (idx >= 8) r = s4 + 1.0f;
        return r;
    };
    const float tm2 = Tat(ell - 2);
    const float tm1 = Tat(ell - 1);
    const float t0  = Tat(ell + 0);
    const float tp1 = Tat(ell + 1);
    const float tp2 = Tat(ell + 2);
    const float tp3 = Tat(ell + 3);

    float r0, r1, r2, r3;
    float den, w, h0, h1, h2;

    // j=1, n=1: tb=T[ell+1], ta=T[ell]
    den = tp1 - t0;
    w = (den == 0.0f) ? 0.0f : (1.0f / den);
    r0 = w * (tp1 - xn);
    r1 = w * (xn - t0);

    // j=2
    h0 = r0; h1 = r1;
    den = tp1 - tm1;                       // n=1
    w = (den == 0.0f) ? 0.0f : (h0 / den);
    r0 = w * (tp1 - xn);
    r1 = w * (xn - tm1);
    den = tp2 - t0;                        // n=2
    w = (den == 0.0f) ? 0.0f : (h1 / den);
    r1 += w * (tp2 - xn);
    r2 = w * (xn - t0);

    // j=3
    h0 = r0; h1 = r1; h2 = r2;
    den = tp1 - tm2;                       // n=1
    w = (den == 0.0f) ? 0.0f : (h0 / den);
    r0 = w * (tp1 - xn);
    r1 = w * (xn - tm2);
    den = tp2 - tm1;                       // n=2
    w = (den == 0.0f) ? 0.0f : (h1 / den);
    r1 += w * (tp2 - xn);
    r2 = w * (xn - tm1);
    den = tp3 - t0;                        // n=3
    w = (den == 0.0f) ? 0.0f : (h2 / den);
    r2 += w * (tp3 - xn);
    r3 = w * (xn - t0);

    const int sel = 6 - ell;               // res[2k - ell]
    float v = (sel == 0) ? r0 : (sel == 1) ? r1 : (sel == 2) ? r2 : r3;
    if (v != v) v = 0.0f;                  // nan_to_num
    return v;
}

// One thread per PAIR of output elements: gid = n*30 + j -> basis i=2j, 2j+1.
// The pair shares xn, 4/6 knots, 4/6 comparisons; stores are aligned b64.
__global__ __launch_bounds__(256) void bspline_basis(const float* __restrict__ x,
                                                     const float* __restrict__ knots,
                                                     const unsigned* __restrict__ ws,
                                                     float* __restrict__ out) {
    __shared__ float sk[NKNOTS];
    const int t = threadIdx.x;

    // Stage the 64 knots into LDS (async gfx1250 path when available).
#if __has_builtin(__builtin_amdgcn_global_load_async_to_lds_b32) && \
    __has_builtin(__builtin_amdgcn_s_wait_asynccnt)
    if (t < NKNOTS) {
        __builtin_amdgcn_global_load_async_to_lds_b32(
            (__attribute__((address_space(1))) int*)(knots + t),
            (__attribute__((address_space(3))) int*)(&sk[t]),
            0, 0);
    }
    __builtin_amdgcn_s_wait_asynccnt(0);
#else
    if (t < NKNOTS) sk[t] = knots[t];
#endif
    __syncthreads();

    const int gid = blockIdx.x * blockDim.x + t;
    if (gid >= TOTAL_PAIRS) return;

    const int n = gid / PAIRS_PER_POINT;
    const int i = (gid - n * PAIRS_PER_POINT) * 2;   // even basis index

    const float mn = f32_dec(ws[0]);
    const float mx = f32_dec(ws[1]);
    const float xn = (x[n] - mn) / (mx - mn + 1e-8f);

    // knots[i .. i+5] cover seg windows of elements i and i+1 (i+5 <= 63)
    const float k0 = sk[i + 0];
    const float k1 = sk[i + 1];
    const float k2 = sk[i + 2];
    const float k3 = sk[i + 3];
    const float k4 = sk[i + 4];
    const float k5 = sk[i + 5];

    const int c0 = (k0 <= xn), c1 = (k1 <= xn), c2 = (k2 <= xn);
    const int c3 = (k3 <= xn), c4 = (k4 <= xn), c5 = (k5 <= xn);

    float2 v;
    v.x = bspline_eval_elem(k0, k1, k2, k3, k4, xn, c0 + c1 + c2 + c3 + c4);
    v.y = bspline_eval_elem(k1, k2, k3, k4, k5, xn, c1 + c2 + c3 + c4 + c5);

    // out element (n, i) at n*60 + i; pair base is even -> 8B aligned b64 store
    *(float2*)(out + n * NB + i) = v;
}

extern "C" void kernel_launch(void* const* d_in, const int* in_sizes, int n_in,
                              void* d_out, int out_size, void* d_ws, size_t ws_size,
                              hipStream_t stream) {
    const float* x     = (const float*)d_in[0];
    const float* knots = (const float*)d_in[1];
    // d_in[2] = degree (fixed at 3, baked into the kernel)
    float*    out = (float*)d_out;
    unsigned* ws  = (unsigned*)d_ws;
    const int n = in_sizes[0];  // 500000

    bspline_init_ws<<<1, 32, 0, stream>>>(ws);

    int red_blocks = (n + 255) / 256;
    if (red_blocks > 2048) red_blocks = 2048;
    bspline_minmax<<<red_blocks, 256, 0, stream>>>(x, ws, n);

    const int pairs  = n * PAIRS_PER_POINT;
    const int blocks = (pairs + 255) / 256;
    bspline_basis<<<blocks, 256, 0, stream>>>(x, knots, ws, out);
}